// NetS3DISFusion_12807592477203
// MI455X (gfx1250) — compile-verified
//
#include <hip/hip_runtime.h>

typedef float v2f __attribute__((ext_vector_type(2)));
typedef float v8f __attribute__((ext_vector_type(8)));

static constexpr int Bb   = 8;
static constexpr int Nn   = 8192;
static constexpr int Kk   = 16;      // neighbors
static constexpr int ROWS = Bb * Nn; // 65536 points
static constexpr int PPB  = 16;      // points per block (one wave32 per block)

// ---------------------------------------------------------------------------
// fp32 WMMA: D(16x16 f32) = A(16x4 f32) * B(4x16 f32) + C
// A layout (ISA 7.12.2): lane 0-15 rows M=0..15 hold K={kb,kb+1}; lanes 16-31
// hold K={kb+2,kb+3}.  B mirrors: lane-half selects K pair, VGPR within pair,
// N = lane&15.  C/D: VGPR r -> row r (lanes 0-15) / r+8 (lanes 16-31).
// ---------------------------------------------------------------------------
__device__ __forceinline__ v8f wmma_f32(v2f a, v2f b, v8f c) {
  return __builtin_amdgcn_wmma_f32_16x16x4_f32(false, a, false, b, (short)0, c,
                                               false, false);
}

__device__ __forceinline__ v2f fragA(const float* base, int stride, int lane, int kb) {
  const int m  = lane & 15;
  const int kh = (lane >> 4) << 1;
  v2f a;
  a.x = base[m * stride + kb + kh];
  a.y = base[m * stride + kb + kh + 1];
  return a;
}
__device__ __forceinline__ v2f fragB(const float* base, int stride, int lane, int kb, int n0) {
  const int n  = n0 + (lane & 15);
  const int kh = (lane >> 4) << 1;
  v2f b;
  b.x = base[(kb + kh) * stride + n];
  b.y = base[(kb + kh + 1) * stride + n];
  return b;
}

// 3-layer kernel-point MLP for one point: d0[16][48] -> d3[16][16], relu+bias.
__device__ __forceinline__ void run_mlp(const float* w1, const float* b1,
                                        const float* w2, const float* b2,
                                        const float* w3, const float* b3,
                                        const float* d0, float* d1, float* d2,
                                        float* d3, int d3s) {
  const int lane = threadIdx.x;
  const int m0   = (lane >> 4) << 3;
  const int col  = lane & 15;
  // layer 1: [16,48] @ [48,32]
#pragma unroll
  for (int nt = 0; nt < 2; ++nt) {
    v8f acc = {};
#pragma unroll
    for (int kb = 0; kb < 48; kb += 4)
      acc = wmma_f32(fragA(d0, 49, lane, kb), fragB(w1, 33, lane, kb, nt * 16), acc);
    const float bv = b1[nt * 16 + col];
#pragma unroll
    for (int r = 0; r < 8; ++r)
      d1[(m0 + r) * 33 + nt * 16 + col] = fmaxf(acc[r] + bv, 0.f);
  }
  __syncthreads();
  // layer 2: [16,32] @ [32,16]
  {
    v8f acc = {};
#pragma unroll
    for (int kb = 0; kb < 32; kb += 4)
      acc = wmma_f32(fragA(d1, 33, lane, kb), fragB(w2, 17, lane, kb, 0), acc);
    const float bv = b2[col];
#pragma unroll
    for (int r = 0; r < 8; ++r)
      d2[(m0 + r) * 17 + col] = fmaxf(acc[r] + bv, 0.f);
  }
  __syncthreads();
  // layer 3: [16,16] @ [16,16]
  {
    v8f acc = {};
#pragma unroll
    for (int kb = 0; kb < 16; kb += 4)
      acc = wmma_f32(fragA(d2, 17, lane, kb), fragB(w3, 17, lane, kb, 0), acc);
    const float bv = b3[col];
#pragma unroll
    for (int r = 0; r < 8; ++r)
      d3[(m0 + r) * d3s + col] = fmaxf(acc[r] + bv, 0.f);
  }
}

// ---------------------------------------------------------------------------
// PtConv1: x = concat(f1,f2) [.,6] -> x1_raw [.,96]   (one wave, 16 points)
// ---------------------------------------------------------------------------
__global__ __launch_bounds__(32) void ptconv1_kernel(
    const float* __restrict__ f1, const float* __restrict__ f2,
    const float* __restrict__ pts, const int* __restrict__ idx,
    const float* __restrict__ cent,
    const float* __restrict__ l1w, const float* __restrict__ l1b,
    const float* __restrict__ l2w, const float* __restrict__ l2b,
    const float* __restrict__ l3w, const float* __restrict__ l3b,
    const float* __restrict__ Wmix,  // [96][96] flattened (cin*16+c, cout)
    float* __restrict__ x1_raw) {
  __shared__ float s_l1w[48][33];
  __shared__ float s_l2w[32][17];
  __shared__ float s_l3w[16][17];
  __shared__ float s_l1b[32];
  __shared__ float s_l2b[16];
  __shared__ float s_l3b[16];
  __shared__ float s_cent[48];
  __shared__ float s_d0[16][49];
  __shared__ float s_d1[16][33];
  __shared__ float s_d2[16][17];
  __shared__ float s_d3[16][17];
  __shared__ float s_ft[16][8];     // gathered feats [k][6]
  __shared__ float s_agg[16][100];  // [point][96] padded

  const int lane = threadIdx.x;
  const int p0   = blockIdx.x * PPB;
  const int bi   = p0 / Nn;  // PPB divides Nn -> block never spans batches

  for (int i = lane; i < 48 * 32; i += 32) s_l1w[i >> 5][i & 31] = l1w[i];
  for (int i = lane; i < 32 * 16; i += 32) s_l2w[i >> 4][i & 15] = l2w[i];
  for (int i = lane; i < 16 * 16; i += 32) s_l3w[i >> 4][i & 15] = l3w[i];
  s_l1b[lane] = l1b[lane];
  if (lane < 16) { s_l2b[lane] = l2b[lane]; s_l3b[lane] = l3b[lane]; }
  for (int i = lane; i < 48; i += 32) s_cent[i] = cent[i];
  __syncthreads();

  for (int p = 0; p < PPB; ++p) {
    const int gp = p0 + p;
    const float cx = pts[gp * 3 + 0], cy = pts[gp * 3 + 1], cz = pts[gp * 3 + 2];
    float dx = 0.f, dy = 0.f, dz = 0.f, n2 = 0.f;
    int gidx = 0;
    if (lane < Kk) {
      const int nbr = idx[gp * Kk + lane];
      gidx = bi * Nn + nbr;
      dx = pts[gidx * 3 + 0] - cx;
      dy = pts[gidx * 3 + 1] - cy;
      dz = pts[gidx * 3 + 2] - cz;
      n2 = dx * dx + dy * dy + dz * dz;
    }
    float mx = n2;
    for (int m = 16; m >= 1; m >>= 1) mx = fmaxf(mx, __shfl_xor(mx, m, 32));
    float maxi = sqrtf(mx);
    if (maxi == 0.f) maxi = 1.f;
    const float inv = 1.f / maxi;
    if (lane < Kk) {
      s_ft[lane][0] = f1[gidx * 3 + 0];
      s_ft[lane][1] = f1[gidx * 3 + 1];
      s_ft[lane][2] = f1[gidx * 3 + 2];
      s_ft[lane][3] = f2[gidx * 3 + 0];
      s_ft[lane][4] = f2[gidx * 3 + 1];
      s_ft[lane][5] = f2[gidx * 3 + 2];
      const float nx = dx * inv, ny = dy * inv, nz = dz * inv;
#pragma unroll
      for (int c = 0; c < 16; ++c) {
        s_d0[lane][c]      = nx - s_cent[c];
        s_d0[lane][16 + c] = ny - s_cent[16 + c];
        s_d0[lane][32 + c] = nz - s_cent[32 + c];
      }
    }
    __syncthreads();
    run_mlp(&s_l1w[0][0], s_l1b, &s_l2w[0][0], s_l2b, &s_l3w[0][0], s_l3b,
            &s_d0[0][0], &s_d1[0][0], &s_d2[0][0], &s_d3[0][0], 17);
    __syncthreads();
    // aggregate (Cin=6, small): agg[cin*16+c] = sum_k ft[k][cin]*d3[k][c]
    for (int i = lane; i < 96; i += 32) {
      const int cin = i >> 4, c = i & 15;
      float s = 0.f;
#pragma unroll
      for (int k = 0; k < Kk; ++k) s = fmaf(s_ft[k][cin], s_d3[k][c], s);
      s_agg[p][i] = s;
    }
    __syncthreads();
  }

  // final mix for the 16 points: [16,96] @ Wmix[96][96], scaled by 1/K
  const int m0  = (lane >> 4) << 3;
  const int col = lane & 15;
  const int kh  = (lane >> 4) << 1;
  for (int nt = 0; nt < 6; ++nt) {
    v8f acc = {};
    for (int kb = 0; kb < 96; kb += 4) {
      v2f a = fragA(&s_agg[0][0], 100, lane, kb);
      v2f b;
      b.x = Wmix[(kb + kh) * 96 + nt * 16 + col];
      b.y = Wmix[(kb + kh + 1) * 96 + nt * 16 + col];
      acc = wmma_f32(a, b, acc);
    }
#pragma unroll
    for (int r = 0; r < 8; ++r)
      x1_raw[(size_t)(p0 + m0 + r) * 96 + nt * 16 + col] = acc[r] * (1.f / 16.f);
  }
}

// ---------------------------------------------------------------------------
// BN training-mode stats -> per-channel scale/shift (one block per channel)
// ---------------------------------------------------------------------------
__global__ __launch_bounds__(256) void bn_stats_kernel(
    const float* __restrict__ x, int rows, int C,
    const float* __restrict__ g, const float* __restrict__ b,
    float* __restrict__ scale, float* __restrict__ shift) {
  __shared__ float ssum[256];
  __shared__ float ssq[256];
  const int c = blockIdx.x;
  const int tid = threadIdx.x;
  float s = 0.f, q = 0.f;
  for (int r = tid; r < rows; r += 256) {
    const float v = x[(size_t)r * C + c];
    s += v;
    q = fmaf(v, v, q);
  }
  ssum[tid] = s;
  ssq[tid] = q;
  __syncthreads();
  for (int off = 128; off > 0; off >>= 1) {
    if (tid < off) { ssum[tid] += ssum[tid + off]; ssq[tid] += ssq[tid + off]; }
    __syncthreads();
  }
  if (tid == 0) {
    const float m   = ssum[0] / (float)rows;
    const float var = ssq[0] / (float)rows - m * m;
    const float sc  = rsqrtf(var + 1e-5f) * g[c];
    scale[c] = sc;
    shift[c] = fmaf(-m, sc, b[c]);
  }
}

// ---------------------------------------------------------------------------
// PtConv2: x1 (BN1+ReLU fused into gather) [.,96] -> x2_raw [.,48]
// ---------------------------------------------------------------------------
__global__ __launch_bounds__(32) void ptconv2_kernel(
    const float* __restrict__ x1_raw, const float* __restrict__ pts,
    const int* __restrict__ idx, const float* __restrict__ cent,
    const float* __restrict__ l1w, const float* __restrict__ l1b,
    const float* __restrict__ l2w, const float* __restrict__ l2b,
    const float* __restrict__ l3w, const float* __restrict__ l3b,
    const float* __restrict__ Wmix,  // [1536][48]
    const float* __restrict__ bnsc, const float* __restrict__ bnsh,
    float* __restrict__ x2_raw) {
  __shared__ float s_l1w[48][33];
  __shared__ float s_l2w[32][17];
  __shared__ float s_l3w[16][17];
  __shared__ float s_l1b[32];
  __shared__ float s_l2b[16];
  __shared__ float s_l3b[16];
  __shared__ float s_cent[48];
  __shared__ float s_sc[96];
  __shared__ float s_sh[96];
  __shared__ float s_d0[16][49];
  __shared__ float s_d1[16][33];
  __shared__ float s_d2[16][17];
  __shared__ float s_d3[16][16 * 17];  // per-point kernel weights [p][k*17+c]
  __shared__ float s_ft[16][17];       // gathered feature tile [k][16ch]
  __shared__ float s_As[16][260];      // final-GEMM A slice [point][256] padded
  __shared__ int   s_nbr[16][16];      // neighbor row ids [p][k]

  const int lane = threadIdx.x;
  const int p0   = blockIdx.x * PPB;
  const int bi   = p0 / Nn;

  for (int i = lane; i < 48 * 32; i += 32) s_l1w[i >> 5][i & 31] = l1w[i];
  for (int i = lane; i < 32 * 16; i += 32) s_l2w[i >> 4][i & 15] = l2w[i];
  for (int i = lane; i < 16 * 16; i += 32) s_l3w[i >> 4][i & 15] = l3w[i];
  s_l1b[lane] = l1b[lane];
  if (lane < 16) { s_l2b[lane] = l2b[lane]; s_l3b[lane] = l3b[lane]; }
  for (int i = lane; i < 48; i += 32) s_cent[i] = cent[i];
  for (int i = lane; i < 96; i += 32) { s_sc[i] = bnsc[i]; s_sh[i] = bnsh[i]; }
  __syncthreads();

  // phase 1: per-point geometry MLP -> persistent d3
  for (int p = 0; p < PPB; ++p) {
    const int gp = p0 + p;
    const float cx = pts[gp * 3 + 0], cy = pts[gp * 3 + 1], cz = pts[gp * 3 + 2];
    float dx = 0.f, dy = 0.f, dz = 0.f, n2 = 0.f;
    if (lane < Kk) {
      const int nbr  = idx[gp * Kk + lane];
      const int gidx = bi * Nn + nbr;
      s_nbr[p][lane] = gidx;
      dx = pts[gidx * 3 + 0] - cx;
      dy = pts[gidx * 3 + 1] - cy;
      dz = pts[gidx * 3 + 2] - cz;
      n2 = dx * dx + dy * dy + dz * dz;
    }
    float mx = n2;
    for (int m = 16; m >= 1; m >>= 1) mx = fmaxf(mx, __shfl_xor(mx, m, 32));
    float maxi = sqrtf(mx);
    if (maxi == 0.f) maxi = 1.f;
    const float inv = 1.f / maxi;
    if (lane < Kk) {
      const float nx = dx * inv, ny = dy * inv, nz = dz * inv;
#pragma unroll
      for (int c = 0; c < 16; ++c) {
        s_d0[lane][c]      = nx - s_cent[c];
        s_d0[lane][16 + c] = ny - s_cent[16 + c];
        s_d0[lane][32 + c] = nz - s_cent[32 + c];
      }
    }
    __syncthreads();
    run_mlp(&s_l1w[0][0], s_l1b, &s_l2w[0][0], s_l2b, &s_l3w[0][0], s_l3b,
            &s_d0[0][0], &s_d1[0][0], &s_d2[0][0], &s_d3[p][0], 17);
    __syncthreads();
  }

  // phase 2: stream 6 cin-tiles; agg WMMA per point feeds the big final GEMM
  v8f accO[3] = {{}, {}, {}};
  const int m0  = (lane >> 4) << 3;
  const int col = lane & 15;
  const int kh  = (lane >> 4) << 1;
  for (int t = 0; t < 6; ++t) {
    for (int p = 0; p < PPB; ++p) {
      // gather 16 neighbors x 16 channels with fused BN1+ReLU (L2-resident)
      for (int i = lane; i < 256; i += 32) {
        const int k = i >> 4, ch = i & 15;
        const int cgl = t * 16 + ch;
        const float v = x1_raw[(size_t)s_nbr[p][k] * 96 + cgl];
        s_ft[k][ch] = fmaxf(fmaf(v, s_sc[cgl], s_sh[cgl]), 0.f);
      }
      __syncthreads();
      // agg[cin_local, c] = sum_k ft[k][cin_local] * d3[k][c]  (A = ft^T)
      v8f agg = {};
#pragma unroll
      for (int kb = 0; kb < 16; kb += 4) {
        v2f a;
        a.x = s_ft[kb + kh][col];
        a.y = s_ft[kb + kh + 1][col];
        v2f b;
        b.x = s_d3[p][(kb + kh) * 17 + col];
        b.y = s_d3[p][(kb + kh + 1) * 17 + col];
        agg = wmma_f32(a, b, agg);
      }
#pragma unroll
      for (int r = 0; r < 8; ++r)
        s_As[p][(m0 + r) * 16 + col] = agg[r];
      __syncthreads();
    }
    // consume the 256-wide K slice of [16 pts,1536]@[1536,48]
#pragma unroll
    for (int nt = 0; nt < 3; ++nt) {
      v8f acc = accO[nt];
      for (int kb = 0; kb < 256; kb += 4) {
        v2f a = fragA(&s_As[0][0], 260, lane, kb);
        v2f b;
        b.x = Wmix[(size_t)(t * 256 + kb + kh) * 48 + nt * 16 + col];
        b.y = Wmix[(size_t)(t * 256 + kb + kh + 1) * 48 + nt * 16 + col];
        acc = wmma_f32(a, b, acc);
      }
      accO[nt] = acc;
    }
    __syncthreads();
  }
#pragma unroll
  for (int r = 0; r < 8; ++r) {
    const size_t row = (size_t)(p0 + m0 + r) * 48;
    x2_raw[row + 0 + col]  = accO[0][r] * (1.f / 16.f);
    x2_raw[row + 16 + col] = accO[1][r] * (1.f / 16.f);
    x2_raw[row + 32 + col] = accO[2][r] * (1.f / 16.f);
  }
}

// ---------------------------------------------------------------------------
// BN2+ReLU + concat(x2,out1,out2) @ fc_w + fc_b  ->  [rows,13]
// ---------------------------------------------------------------------------
__global__ __launch_bounds__(256) void fuse_out_kernel(
    const float* __restrict__ x2_raw, const float* __restrict__ sc,
    const float* __restrict__ sh, const float* __restrict__ o1,
    const float* __restrict__ o2, const float* __restrict__ fcw,
    const float* __restrict__ fcb, float* __restrict__ out, int rows) {
  const int i = blockIdx.x * 256 + threadIdx.x;
  if (i >= rows * 13) return;
  const int p = i / 13, co = i % 13;
  float s = fcb[co];
#pragma unroll
  for (int j = 0; j < 48; ++j) {
    const float v = fmaxf(fmaf(x2_raw[(size_t)p * 48 + j], sc[j], sh[j]), 0.f);
    s = fmaf(v, fcw[j * 13 + co], s);
  }
#pragma unroll
  for (int j = 0; j < 13; ++j) s = fmaf(o1[(size_t)p * 13 + j], fcw[(48 + j) * 13 + co], s);
#pragma unroll
  for (int j = 0; j < 13; ++j) s = fmaf(o2[(size_t)p * 13 + j], fcw[(61 + j) * 13 + co], s);
  out[i] = s;
}

// ---------------------------------------------------------------------------
extern "C" void kernel_launch(void* const* d_in, const int* in_sizes, int n_in,
                              void* d_out, int out_size, void* d_ws, size_t ws_size,
                              hipStream_t stream) {
  const float* out1  = (const float*)d_in[0];
  const float* out2  = (const float*)d_in[1];
  const float* f1    = (const float*)d_in[2];
  const float* f2    = (const float*)d_in[3];
  const float* pts   = (const float*)d_in[4];
  const int*   idx   = (const int*)d_in[5];
  const float* c1c   = (const float*)d_in[6];
  const float* c1l1w = (const float*)d_in[7];
  const float* c1l1b = (const float*)d_in[8];
  const float* c1l2w = (const float*)d_in[9];
  const float* c1l2b = (const float*)d_in[10];
  const float* c1l3w = (const float*)d_in[11];
  const float* c1l3b = (const float*)d_in[12];
  const float* c1w   = (const float*)d_in[13];
  const float* c2c   = (const float*)d_in[14];
  const float* c2l1w = (const float*)d_in[15];
  const float* c2l1b = (const float*)d_in[16];
  const float* c2l2w = (const float*)d_in[17];
  const float* c2l2b = (const float*)d_in[18];
  const float* c2l3w = (const float*)d_in[19];
  const float* c2l3b = (const float*)d_in[20];
  const float* c2w   = (const float*)d_in[21];
  const float* bn1g  = (const float*)d_in[22];
  const float* bn1b  = (const float*)d_in[23];
  const float* bn2g  = (const float*)d_in[24];
  const float* bn2b  = (const float*)d_in[25];
  const float* fcw   = (const float*)d_in[26];
  const float* fcb   = (const float*)d_in[27];

  float* x1_raw = (float*)d_ws;                       // 65536*96
  float* x2_raw = x1_raw + (size_t)ROWS * 96;         // 65536*48
  float* bn1sc  = x2_raw + (size_t)ROWS * 48;
  float* bn1sh  = bn1sc + 96;
  float* bn2sc  = bn1sh + 96;
  float* bn2sh  = bn2sc + 48;

  ptconv1_kernel<<<ROWS / PPB, 32, 0, stream>>>(f1, f2, pts, idx, c1c, c1l1w,
                                                c1l1b, c1l2w, c1l2b, c1l3w,
                                                c1l3b, c1w, x1_raw);
  bn_stats_kernel<<<96, 256, 0, stream>>>(x1_raw, ROWS, 96, bn1g, bn1b, bn1sc, bn1sh);
  ptconv2_kernel<<<ROWS / PPB, 32, 0, stream>>>(x1_raw, pts, idx, c2c, c2l1w,
                                                c2l1b, c2l2w, c2l2b, c2l3w,
                                                c2l3b, c2w, bn1sc, bn1sh, x2_raw);
  bn_stats_kernel<<<48, 256, 0, stream>>>(x2_raw, ROWS, 48, bn2g, bn2b, bn2sc, bn2sh);
  fuse_out_kernel<<<(ROWS * 13 + 255) / 256, 256, 0, stream>>>(
      x2_raw, bn2sc, bn2sh, out1, out2, fcw, fcb, (float*)d_out, ROWS);
}